// EdgeGatedGraphConv_64424509440774
// MI455X (gfx1250) — compile-verified
//
#include <hip/hip_runtime.h>

typedef __attribute__((ext_vector_type(2))) float v2f;
typedef __attribute__((ext_vector_type(8))) float v8f;

#define N_NODES 50000
#define N_EDGES 800000
#define DF 128
#define BN_EPS 1e-5f

// ---------------------------------------------------------------------------
// GEMM: Out[rows,128] = X[rows,128] @ W^T + bias, W is [128,128] row-major.
// One wave computes a 16-row x 128-col strip: 8 accumulators (16x16 tiles),
// k-loop in steps of 4 using V_WMMA_F32_16X16X4_F32. A-frags reused 8x.
// rows must be a multiple of 16 (50000 and 800000 both are).
// ---------------------------------------------------------------------------
__global__ __launch_bounds__(256) void wmma_gemm_xwT(
    const float* __restrict__ X, const float* __restrict__ W,
    const float* __restrict__ bias, float* __restrict__ Out, int rows)
{
    const int lane = threadIdx.x & 31;
    const int wave = threadIdx.x >> 5;
    const int tile = blockIdx.x * 8 + wave;       // 16-row strip index (wave-uniform)
    if (tile * 16 >= rows) return;                // wave-uniform exit: EXEC stays all-ones
    const int row0 = tile * 16;
    const int half = lane >> 4;                   // 0: K=0,1   1: K=2,3
    const int lm   = lane & 15;

    const float* xrow = X + (size_t)(row0 + lm) * DF + half * 2;

    v8f acc[8];
#pragma unroll
    for (int t = 0; t < 8; ++t)
        acc[t] = v8f{0.f, 0.f, 0.f, 0.f, 0.f, 0.f, 0.f, 0.f};

    for (int kk = 0; kk < DF; kk += 4) {
        v2f a = *(const v2f*)(xrow + kk);
#pragma unroll
        for (int t = 0; t < 8; ++t) {
            // B[k][n] = W[n][k]; lane lm -> col t*16+lm, halves pick k pairs
            v2f b = *(const v2f*)(W + (size_t)(t * 16 + lm) * DF + kk + half * 2);
            acc[t] = __builtin_amdgcn_wmma_f32_16x16x4_f32(
                false, a, false, b, (short)0, acc[t], false, false);
        }
    }

#pragma unroll
    for (int t = 0; t < 8; ++t) {
        const float bv = bias[t * 16 + lm];
#pragma unroll
        for (int r = 0; r < 8; ++r) {
            const int row = row0 + half * 8 + r;  // C/D layout: VGPR r -> M = half*8+r
            Out[(size_t)row * DF + t * 16 + lm] = acc[t][r] + bv;
        }
    }
}

// ---------------------------------------------------------------------------
// Edge pass: one wave per edge (lane handles 4 consecutive features).
// y = Psg[i] + Pdg[j] + y_proj (in-place in d_out's y region); sigma = sigmoid(y);
// m = Pdu[j] * sigma; atomic segment-sums into sum_m / sum_sigma;
// fused edge-BN column stats (register accum -> LDS -> 2 atomics per column).
// ---------------------------------------------------------------------------
__global__ __launch_bounds__(256) void edge_gate_kernel(
    const int* __restrict__ edge_index,
    const float* __restrict__ Psg, const float* __restrict__ Pdg,
    const float* __restrict__ Pdu,
    float* __restrict__ Y,               // [E,128] in: proj, out: y (pre-BN)
    float* __restrict__ sum_m, float* __restrict__ sum_sigma,
    float* __restrict__ edge_stats,      // [256]: sum[128] | sumsq[128]
    int E)
{
    __shared__ float ls[8][DF];
    __shared__ float lq[8][DF];
    const int lane = threadIdx.x & 31;
    const int wave = threadIdx.x >> 5;
    const int c0   = lane * 4;
    const int stride = gridDim.x * 8;

    float s0 = 0.f, s1 = 0.f, s2 = 0.f, s3 = 0.f;
    float q0 = 0.f, q1 = 0.f, q2 = 0.f, q3 = 0.f;

    for (int e = blockIdx.x * 8 + wave; e < E; e += stride) {
        const int i = edge_index[e];
        const int j = edge_index[E + e];
        const float4 yp = *(const float4*)(Y   + (size_t)e * DF + c0);
        const float4 pa = *(const float4*)(Psg + (size_t)i * DF + c0);
        const float4 pb = *(const float4*)(Pdg + (size_t)j * DF + c0);
        const float4 du = *(const float4*)(Pdu + (size_t)j * DF + c0);

        const float y0 = pa.x + pb.x + yp.x;
        const float y1 = pa.y + pb.y + yp.y;
        const float y2 = pa.z + pb.z + yp.z;
        const float y3 = pa.w + pb.w + yp.w;

        const float g0 = 1.f / (1.f + __expf(-y0));
        const float g1 = 1.f / (1.f + __expf(-y1));
        const float g2 = 1.f / (1.f + __expf(-y2));
        const float g3 = 1.f / (1.f + __expf(-y3));

        float4 yo; yo.x = y0; yo.y = y1; yo.z = y2; yo.w = y3;
        *(float4*)(Y + (size_t)e * DF + c0) = yo;

        float* pm = sum_m     + (size_t)i * DF + c0;
        float* ps = sum_sigma + (size_t)i * DF + c0;
        atomicAdd(pm + 0, du.x * g0); atomicAdd(pm + 1, du.y * g1);
        atomicAdd(pm + 2, du.z * g2); atomicAdd(pm + 3, du.w * g3);
        atomicAdd(ps + 0, g0);        atomicAdd(ps + 1, g1);
        atomicAdd(ps + 2, g2);        atomicAdd(ps + 3, g3);

        s0 += y0; s1 += y1; s2 += y2; s3 += y3;
        q0 += y0 * y0; q1 += y1 * y1; q2 += y2 * y2; q3 += y3 * y3;
    }

    ls[wave][c0 + 0] = s0; ls[wave][c0 + 1] = s1;
    ls[wave][c0 + 2] = s2; ls[wave][c0 + 3] = s3;
    lq[wave][c0 + 0] = q0; lq[wave][c0 + 1] = q1;
    lq[wave][c0 + 2] = q2; lq[wave][c0 + 3] = q3;
    __syncthreads();
    if (threadIdx.x < DF) {
        float ts = 0.f, tq = 0.f;
#pragma unroll
        for (int w = 0; w < 8; ++w) { ts += ls[w][threadIdx.x]; tq += lq[w][threadIdx.x]; }
        atomicAdd(&edge_stats[threadIdx.x], ts);
        atomicAdd(&edge_stats[DF + threadIdx.x], tq);
    }
}

// ---------------------------------------------------------------------------
// Node pass: x_pre = Psu + sum_m / (sum_sigma + 1e-6), fused node-BN stats.
// One wave per node row.
// ---------------------------------------------------------------------------
__global__ __launch_bounds__(256) void node_pre_kernel(
    const float* __restrict__ Psu, const float* __restrict__ sum_m,
    const float* __restrict__ sum_sigma,
    float* __restrict__ Xpre, float* __restrict__ node_stats, int N)
{
    __shared__ float ls[8][DF];
    __shared__ float lq[8][DF];
    const int lane = threadIdx.x & 31;
    const int wave = threadIdx.x >> 5;
    const int c0   = lane * 4;
    const int stride = gridDim.x * 8;

    float s0 = 0.f, s1 = 0.f, s2 = 0.f, s3 = 0.f;
    float q0 = 0.f, q1 = 0.f, q2 = 0.f, q3 = 0.f;

    for (int n = blockIdx.x * 8 + wave; n < N; n += stride) {
        const float4 pu = *(const float4*)(Psu       + (size_t)n * DF + c0);
        const float4 sm = *(const float4*)(sum_m     + (size_t)n * DF + c0);
        const float4 ss = *(const float4*)(sum_sigma + (size_t)n * DF + c0);
        const float x0 = pu.x + sm.x / (ss.x + 1e-6f);
        const float x1 = pu.y + sm.y / (ss.y + 1e-6f);
        const float x2 = pu.z + sm.z / (ss.z + 1e-6f);
        const float x3 = pu.w + sm.w / (ss.w + 1e-6f);
        float4 xo; xo.x = x0; xo.y = x1; xo.z = x2; xo.w = x3;
        *(float4*)(Xpre + (size_t)n * DF + c0) = xo;
        s0 += x0; s1 += x1; s2 += x2; s3 += x3;
        q0 += x0 * x0; q1 += x1 * x1; q2 += x2 * x2; q3 += x3 * x3;
    }

    ls[wave][c0 + 0] = s0; ls[wave][c0 + 1] = s1;
    ls[wave][c0 + 2] = s2; ls[wave][c0 + 3] = s3;
    lq[wave][c0 + 0] = q0; lq[wave][c0 + 1] = q1;
    lq[wave][c0 + 2] = q2; lq[wave][c0 + 3] = q3;
    __syncthreads();
    if (threadIdx.x < DF) {
        float ts = 0.f, tq = 0.f;
#pragma unroll
        for (int w = 0; w < 8; ++w) { ts += ls[w][threadIdx.x]; tq += lq[w][threadIdx.x]; }
        atomicAdd(&node_stats[threadIdx.x], ts);
        atomicAdd(&node_stats[DF + threadIdx.x], tq);
    }
}

// ---------------------------------------------------------------------------
// BN (training-mode batch stats) + SiLU + residual, in place. float4 per thread.
// ---------------------------------------------------------------------------
__global__ __launch_bounds__(256) void bn_silu_res_kernel(
    float* __restrict__ V,               // [rows,128] pre-BN in, final out
    const float* __restrict__ resid,     // [rows,128]
    const float* __restrict__ gamma, const float* __restrict__ beta,
    const float* __restrict__ stats,     // [256]: sum | sumsq
    float inv_count, long long n4)       // n4 = rows * 32 (float4 units)
{
    long long idx = (long long)blockIdx.x * blockDim.x + threadIdx.x;
    if (idx >= n4) return;
    const int c0 = (int)(idx & 31) * 4;

    const float4 v = *(const float4*)(V + idx * 4);
    const float4 r = *(const float4*)(resid + idx * 4);
    float out[4];
    const float vin[4] = {v.x, v.y, v.z, v.w};
    const float rin[4] = {r.x, r.y, r.z, r.w};
#pragma unroll
    for (int k = 0; k < 4; ++k) {
        const int c = c0 + k;
        const float mean = stats[c] * inv_count;
        const float var  = stats[DF + c] * inv_count - mean * mean;
        const float z = (vin[k] - mean) * rsqrtf(var + BN_EPS) * gamma[c] + beta[c];
        out[k] = rin[k] + z / (1.f + __expf(-z));     // resid + silu(z)
    }
    float4 o; o.x = out[0]; o.y = out[1]; o.z = out[2]; o.w = out[3];
    *(float4*)(V + idx * 4) = o;
}

// ---------------------------------------------------------------------------
extern "C" void kernel_launch(void* const* d_in, const int* in_sizes, int n_in,
                              void* d_out, int out_size, void* d_ws, size_t ws_size,
                              hipStream_t stream)
{
    const float* node_feats = (const float*)d_in[0];
    const float* edge_feats = (const float*)d_in[1];
    const int*   edge_index = (const int*)d_in[2];
    const float* W_sg = (const float*)d_in[3];  const float* b_sg = (const float*)d_in[4];
    const float* W_dg = (const float*)d_in[5];  const float* b_dg = (const float*)d_in[6];
    const float* W_eg = (const float*)d_in[7];  const float* b_eg = (const float*)d_in[8];
    const float* W_su = (const float*)d_in[9];  const float* b_su = (const float*)d_in[10];
    const float* W_du = (const float*)d_in[11]; const float* b_du = (const float*)d_in[12];
    const float* gamma_n = (const float*)d_in[13]; const float* beta_n = (const float*)d_in[14];
    const float* gamma_e = (const float*)d_in[15]; const float* beta_e = (const float*)d_in[16];

    const size_t ND = (size_t)N_NODES * DF;

    float* out_x = (float*)d_out;            // [N,128]
    float* out_y = out_x + ND;               // [E,128]: y_proj -> y_pre -> y_final

    float* ws        = (float*)d_ws;
    float* P_sg      = ws;
    float* P_dg      = ws + 1 * ND;
    float* P_du      = ws + 2 * ND;
    float* P_su      = ws + 3 * ND;
    float* sum_m     = ws + 4 * ND;
    float* sum_sigma = ws + 5 * ND;
    float* node_stats = ws + 6 * ND;         // 256 floats
    float* edge_stats = node_stats + 256;    // 256 floats

    // Zero accumulators + stats (graph-capture-legal async memset)
    hipMemsetAsync(sum_m, 0, (2 * ND + 512) * sizeof(float), stream);

    // GEMMs: one wave per 16-row strip, 8 strips per 256-thread block
    const int nblk = (N_NODES / 16 + 7) / 8;   // 391
    const int eblk = (N_EDGES / 16 + 7) / 8;   // 6250
    wmma_gemm_xwT<<<nblk, 256, 0, stream>>>(node_feats, W_sg, b_sg, P_sg, N_NODES);
    wmma_gemm_xwT<<<nblk, 256, 0, stream>>>(node_feats, W_dg, b_dg, P_dg, N_NODES);
    wmma_gemm_xwT<<<nblk, 256, 0, stream>>>(node_feats, W_du, b_du, P_du, N_NODES);
    wmma_gemm_xwT<<<nblk, 256, 0, stream>>>(node_feats, W_su, b_su, P_su, N_NODES);
    wmma_gemm_xwT<<<eblk, 256, 0, stream>>>(edge_feats, W_eg, b_eg, out_y, N_EDGES);

    // Edge gating + segment-sum atomics + edge-BN stats
    edge_gate_kernel<<<2048, 256, 0, stream>>>(edge_index, P_sg, P_dg, P_du,
                                               out_y, sum_m, sum_sigma, edge_stats, N_EDGES);

    // x_pre + node-BN stats
    node_pre_kernel<<<(N_NODES + 7) / 8, 256, 0, stream>>>(P_su, sum_m, sum_sigma,
                                                           out_x, node_stats, N_NODES);

    // BN + SiLU + residual, in place
    const long long n4_nodes = (long long)N_NODES * (DF / 4);
    const long long n4_edges = (long long)N_EDGES * (DF / 4);
    bn_silu_res_kernel<<<(int)((n4_nodes + 255) / 256), 256, 0, stream>>>(
        out_x, node_feats, gamma_n, beta_n, node_stats, 1.f / N_NODES, n4_nodes);
    bn_silu_res_kernel<<<(int)((n4_edges + 255) / 256), 256, 0, stream>>>(
        out_y, edge_feats, gamma_e, beta_e, edge_stats, 1.f / N_EDGES, n4_edges);
}